// TFFeatureGen_11579231830047
// MI455X (gfx1250) — compile-verified
//
#include <hip/hip_runtime.h>
#include <hip/hip_bf16.h>

// ---- problem constants (from reference) ----
#define T_FRAMES   16384
#define N_LM       543
#define L_OUT      104          // 4 group means + 100 gathered landmarks
#define NCOL       312          // L_OUT * 3
#define SEGMENTS   8
#define SEG_ROWS   2048         // T_FRAMES / SEGMENTS
#define TILES      20           // ceil(312/16)
#define CHUNKS     8            // row chunks per segment (256 rows each)
#define CHUNK_ROWS 256
#define FEAT_BLK   624          // 2 * NCOL
#define RESIZE_T   64

typedef float v2f __attribute__((ext_vector_type(2)));
typedef float v8f __attribute__((ext_vector_type(8)));

__device__ __forceinline__ float fin_or0(float v) { return (v - v == 0.0f) ? v : 0.0f; } // isfinite ? v : 0

// ---------------------------------------------------------------------------
// K1: build x[t, 104, 3] = 4 group nan-means + 100-landmark gather.
// One block (256 threads) per time step; row of 1629 floats is read coalesced.
// ---------------------------------------------------------------------------
__global__ void k1_build_x(const float* __restrict__ xin,
                           const int*   __restrict__ pl,
                           float*       __restrict__ X) {
  const int t   = blockIdx.x;
  const int tid = threadIdx.x;
  __shared__ float s_sum[12];
  __shared__ float s_cnt[12];
  if (tid < 12) { s_sum[tid] = 0.0f; s_cnt[tid] = 0.0f; }
  __syncthreads();

  const float* __restrict__ row = xin + (size_t)t * (N_LM * 3);

  const int gs[4] = {0, 468, 489, 522};
  const int gl[4] = {468, 21, 33, 21};
#pragma unroll
  for (int g = 0; g < 4; ++g) {
    float sx = 0.f, sy = 0.f, sz = 0.f, nx = 0.f, ny = 0.f, nz = 0.f;
    for (int l = gs[g] + tid; l < gs[g] + gl[g]; l += 256) {
      float vx = row[l * 3 + 0], vy = row[l * 3 + 1], vz = row[l * 3 + 2];
      if (vx == vx) { sx += vx; nx += 1.f; }
      if (vy == vy) { sy += vy; ny += 1.f; }
      if (vz == vz) { sz += vz; nz += 1.f; }
    }
    if (tid < gl[g]) {
      atomicAdd(&s_sum[g * 3 + 0], sx); atomicAdd(&s_cnt[g * 3 + 0], nx);
      atomicAdd(&s_sum[g * 3 + 1], sy); atomicAdd(&s_cnt[g * 3 + 1], ny);
      atomicAdd(&s_sum[g * 3 + 2], sz); atomicAdd(&s_cnt[g * 3 + 2], nz);
    }
  }
  __syncthreads();

  float* __restrict__ xrow = X + (size_t)t * NCOL;
  if (tid < 12) xrow[tid] = s_sum[tid] / s_cnt[tid];     // 0/0 -> NaN matches nan_mean of empty
  if (tid < 100) {
    int idx = pl[tid];
    xrow[12 + tid * 3 + 0] = row[idx * 3 + 0];
    xrow[12 + tid * 3 + 1] = row[idx * 3 + 1];
    xrow[12 + tid * 3 + 2] = row[idx * 3 + 2];
  }
}

// ---------------------------------------------------------------------------
// K2: WMMA column reductions. One wave per (tile, segment, row-chunk).
// D = ones[16x4] x B[4x16] + C accumulates 4 rows x 16 cols per v_wmma.
// Three accumulators: sum(x), sum(mask), sum(x^2) over NaN-masked data.
// B layout (32-bit, 4x16): VGPR0 = {K0 lanes0-15, K2 lanes16-31},
//                          VGPR1 = {K1 lanes0-15, K3 lanes16-31}.
// D row 0 = VGPR0 of lanes 0-15.
// Tail columns (312..319) use a CLAMPED in-bounds address so every
// global_load is unconditional (no exec-mask branches in the hot loop);
// their contribution is zeroed in the value/mask selects instead.
// ---------------------------------------------------------------------------
__global__ void k2_wmma_colsums(const float* __restrict__ X,
                                float*       __restrict__ partials) {
  const int tile  = blockIdx.x;   // 0..19
  const int seg   = blockIdx.y;   // 0..7
  const int chunk = blockIdx.z;   // 0..7
  const int lane  = threadIdx.x;  // 0..31
  const int half  = lane >> 4;    // 0 -> K0/K1, 1 -> K2/K3
  const int n     = lane & 15;
  const int c     = tile * 16 + n;
  const bool cok  = (c < NCOL);
  const int csafe = cok ? c : (NCOL - 1);   // always-in-bounds address

  const size_t rowbase = (size_t)(seg * SEG_ROWS + chunk * CHUNK_ROWS);
  const float* __restrict__ Xp = X + rowbase * NCOL + csafe;

  v2f ones; ones.x = 1.0f; ones.y = 1.0f;
  v8f accS = {}; v8f accN = {}; v8f accQ = {};

  for (int r = 0; r < CHUNK_ROWS; r += 4) {
    // unconditional, fully coalesced row loads (rows r+2*half, r+2*half+1)
    float x0 = Xp[(size_t)(r + 2 * half + 0) * NCOL];
    float x1 = Xp[(size_t)(r + 2 * half + 1) * NCOL];
    bool  g0 = (x0 == x0) & cok;            // valid (non-NaN, real column)
    bool  g1 = (x1 == x1) & cok;
    float m0 = g0 ? 1.0f : 0.0f;
    float m1 = g1 ? 1.0f : 0.0f;
    float v0 = g0 ? x0 : 0.0f;
    float v1 = g1 ? x1 : 0.0f;
    v2f bx; bx.x = v0;      bx.y = v1;
    v2f bm; bm.x = m0;      bm.y = m1;
    v2f bq; bq.x = v0 * v0; bq.y = v1 * v1;
    accS = __builtin_amdgcn_wmma_f32_16x16x4_f32(false, ones, false, bx, (short)0, accS, false, false);
    accN = __builtin_amdgcn_wmma_f32_16x16x4_f32(false, ones, false, bm, (short)0, accN, false, false);
    accQ = __builtin_amdgcn_wmma_f32_16x16x4_f32(false, ones, false, bq, (short)0, accQ, false, false);
  }

  if (half == 0 && cok) {
    size_t base = ((((size_t)seg * TILES + tile) * CHUNKS + chunk) * 3) * 16 + n;
    partials[base + 0 * 16] = accS[0];   // D[0][n]
    partials[base + 1 * 16] = accN[0];
    partials[base + 2 * 16] = accQ[0];
  }
}

// ---------------------------------------------------------------------------
// K3: finalize statistics. One thread per column: per-segment mean/std,
// full-sequence mean/std, mean0 for imputation. isfinite->0 guard as in ref.
// ---------------------------------------------------------------------------
__global__ void k3_finalize(const float* __restrict__ partials,
                            float*       __restrict__ out,
                            float*       __restrict__ mean0) {
  const int c = threadIdx.x;
  if (c >= NCOL) return;
  const int tile = c >> 4, n = c & 15;

  float tS = 0.f, tN = 0.f, tQ = 0.f;
  for (int s = 0; s < SEGMENTS; ++s) {
    float S = 0.f, N = 0.f, Q = 0.f;
    for (int ch = 0; ch < CHUNKS; ++ch) {
      size_t base = ((((size_t)s * TILES + tile) * CHUNKS + ch) * 3) * 16 + n;
      S += partials[base + 0 * 16];
      N += partials[base + 1 * 16];
      Q += partials[base + 2 * 16];
    }
    float mean = S / N;
    float var  = fmaxf(Q / N - mean * mean, 0.0f);
    out[s * FEAT_BLK + c]        = fin_or0(mean);
    out[s * FEAT_BLK + NCOL + c] = fin_or0(sqrtf(var));
    tS += S; tN += N; tQ += Q;
  }
  float mean = tS / tN;
  float var  = fmaxf(tQ / tN - mean * mean, 0.0f);
  out[SEGMENTS * FEAT_BLK + c]        = fin_or0(mean);
  out[SEGMENTS * FEAT_BLK + NCOL + c] = fin_or0(sqrtf(var));
  mean0[c] = mean;  // keep NaN so it propagates through the resize, then ->0
}

// ---------------------------------------------------------------------------
// K4: antialiased bilinear (triangle) time resize 16384 -> 64 with NaN
// imputation by mean0. Output center i maps to input 256*i + 127.5; triangle
// half-width 256 -> 512 taps, clipped at edges and weight-normalized.
// ---------------------------------------------------------------------------
__global__ void k4_resize(const float* __restrict__ X,
                          const float* __restrict__ mean0,
                          float*       __restrict__ out) {
  const int c = blockIdx.x * blockDim.x + threadIdx.x;
  const int i = blockIdx.y;
  if (c >= NCOL) return;
  const float m0 = mean0[c];
  const float center = 256.0f * (float)i + 127.5f;
  const int t0 = max(0, 256 * i - 128);
  const int t1 = min(T_FRAMES - 1, 256 * i + 383);
  float acc = 0.f, wsum = 0.f;
  for (int t = t0; t <= t1; ++t) {
    float w = 1.0f - fabsf((float)t - center) * (1.0f / 256.0f);
    float v = X[(size_t)t * NCOL + c];
    v = (v - v == 0.0f) ? v : m0;        // isfinite(x) ? x : mean0
    acc += w * v; wsum += w;
  }
  float r = acc / wsum;
  out[9 * FEAT_BLK + (size_t)i * NCOL + c] = (r == r) ? r : 0.0f;  // isnan -> 0
}

// ---------------------------------------------------------------------------
extern "C" void kernel_launch(void* const* d_in, const int* in_sizes, int n_in,
                              void* d_out, int out_size, void* d_ws, size_t ws_size,
                              hipStream_t stream) {
  const float* x_in = (const float*)d_in[0];
  const int*   pl   = (const int*)d_in[1];
  float* out = (float*)d_out;

  float* X        = (float*)d_ws;                                // 16384*312 floats (~20.4 MB)
  float* partials = X + (size_t)T_FRAMES * NCOL;                 // 8*20*8*3*16 = 61440 floats
  float* mean0    = partials + (size_t)SEGMENTS * TILES * CHUNKS * 3 * 16; // 312 floats

  k1_build_x<<<dim3(T_FRAMES), dim3(256), 0, stream>>>(x_in, pl, X);
  k2_wmma_colsums<<<dim3(TILES, SEGMENTS, CHUNKS), dim3(32), 0, stream>>>(X, partials);
  k3_finalize<<<dim3(1), dim3(320), 0, stream>>>(partials, out, mean0);
  k4_resize<<<dim3(3, RESIZE_T), dim3(128), 0, stream>>>(X, mean0, out);
}